// SuperpointNeuralOperator_23613730193456
// MI455X (gfx1250) — compile-verified
//
#include <hip/hip_runtime.h>
#include <hip/hip_bf16.h>
#include <math.h>

// ---------------------------------------------------------------------------
// SuperpointNeuralOperator fused pipeline for MI455X (gfx1250, wave32, WMMA).
// All matrix work uses V_WMMA_F32_16X16X4_F32 (matches fp32 reference).
// Weights are staged in LDS in a "paired" layout so each WMMA B-fragment is a
// single aligned ds_load_b64 into an adjacent VGPR pair (no mov fix-ups).
// ---------------------------------------------------------------------------

#define N_PTS 65536
#define KNN   16
#define HD    64
#define T_STEPS 3
#define KA    8
#define NSP   512
#define CHUNK 128   // ceil(N/NSP)

typedef __attribute__((ext_vector_type(2))) float v2f;
typedef __attribute__((ext_vector_type(8))) float v8f;

__device__ __forceinline__ v8f wmma4(v2f a, v2f b, v8f c) {
  // D = A(16x4) * B(4x16) + C(16x16), fp32 matrix core
  return __builtin_amdgcn_wmma_f32_16x16x4_f32(false, a, false, b, (short)0, c,
                                               false, false);
}

__device__ __forceinline__ float gelu_exact(float x) {
  return 0.5f * x * (1.0f + erff(x * 0.70710678118654752f));
}
__device__ __forceinline__ float sigmoidf(float x) {
  return 1.0f / (1.0f + expf(-x));
}

// Stage a 64x64 row-major weight into WMMA-B "paired" LDS layout:
//   sWp[ ((kk*2 + h)*64 + c)*2 + lohi ] = W[(kk*4 + 2h + lohi)*64 + c]
// so lane (half h) fetches its (k, k+1) pair for column c with one b64 load.
__device__ __forceinline__ void stage_w_pairs(const float* __restrict__ W,
                                              float* __restrict__ sWp,
                                              int tid, int nthreads) {
  for (int k = tid; k < 4096; k += nthreads) {
    int r = k >> 6, c = k & 63;
    int kk = r >> 2, h = (r >> 1) & 1, lohi = r & 1;
    sWp[((((kk << 1) + h) << 6) + c) * 2 + lohi] = W[k];
  }
}

// B-fragment fetch from paired layout (single aligned ds_load_b64)
__device__ __forceinline__ v2f bfrag(const float* __restrict__ sWp, int kk,
                                     int hi, int n) {
  return *(const v2f*)&sWp[((((kk << 1) + hi) << 6) + n) * 2];
}

// ---------------------------------------------------------------------------
// 1) lift: v = [coords(3) | feat(6)] @ lift_w(9x64) + lift_b
// ---------------------------------------------------------------------------
__global__ void k_lift(const float* __restrict__ coords,
                       const float* __restrict__ feat,
                       const float* __restrict__ w,
                       const float* __restrict__ b,
                       float* __restrict__ v) {
  int g = blockIdx.x * blockDim.x + threadIdx.x;  // N*H threads
  int i = g >> 6, h = g & 63;
  float acc = b[h];
#pragma unroll
  for (int c = 0; c < 3; ++c) acc += coords[(size_t)i * 3 + c] * w[c * 64 + h];
#pragma unroll
  for (int f = 0; f < 6; ++f) acc += feat[(size_t)i * 6 + f] * w[(3 + f) * 64 + h];
  v[g] = acc;
}

// ---------------------------------------------------------------------------
// 2) dual GEMM: vi_all = v @ gvi_w + gvi_b ; vj_all = v @ gvj_w
//    8 waves/block, 16 rows per wave, A-fragments shared across both weights.
// ---------------------------------------------------------------------------
#define DG_WAVES 8
__global__ void k_dual_gemm(const float* __restrict__ v,
                            const float* __restrict__ Wi,
                            const float* __restrict__ bi,
                            const float* __restrict__ Wj,
                            float* __restrict__ outI,
                            float* __restrict__ outJ) {
  __shared__ float sWi[64 * 64];
  __shared__ float sWj[64 * 64];
  int tid = threadIdx.x;
  stage_w_pairs(Wi, sWi, tid, DG_WAVES * 32);
  stage_w_pairs(Wj, sWj, tid, DG_WAVES * 32);
  __syncthreads();

  int wave = tid >> 5, lane = tid & 31;
  int ln = lane & 15, hi = lane >> 4;
  int row0 = (blockIdx.x * DG_WAVES + wave) * 16;

  v2f af[16];  // A fragments: row = row0+ln, k = kk*4 + 2*hi + {0,1}
#pragma unroll
  for (int kk = 0; kk < 16; ++kk)
    af[kk] = *(const v2f*)(v + (size_t)(row0 + ln) * 64 + kk * 4 + 2 * hi);

  v8f accI[4], accJ[4];
#pragma unroll
  for (int nt = 0; nt < 4; ++nt) {
    float bv = bi[nt * 16 + ln];
#pragma unroll
    for (int r = 0; r < 8; ++r) { accI[nt][r] = bv; accJ[nt][r] = 0.f; }
#pragma unroll
    for (int kk = 0; kk < 16; ++kk) {
      accI[nt] = wmma4(af[kk], bfrag(sWi, kk, hi, nt * 16 + ln), accI[nt]);
      accJ[nt] = wmma4(af[kk], bfrag(sWj, kk, hi, nt * 16 + ln), accJ[nt]);
    }
  }
#pragma unroll
  for (int nt = 0; nt < 4; ++nt)
#pragma unroll
    for (int r = 0; r < 8; ++r) {
      size_t off = (size_t)(row0 + r + 8 * hi) * 64 + nt * 16 + ln;
      outI[off] = accI[nt][r];
      outJ[off] = accJ[nt][r];
    }
}

// ---------------------------------------------------------------------------
// 3) fused message + update, one wave per point:
//    g = pos_proj + vi + vj ; h = gelu(g) ; h2 = gelu(h@gm1+b) ;
//    G = sigmoid(h2.gm2+b) ; integral = mean_k(G * v[idx]) ;
//    v' = LN(relu(integral + v@W_w))
// ---------------------------------------------------------------------------
#define MU_WAVES 4
__global__ void k_message_update(
    const float* __restrict__ vold, const float* __restrict__ vi_all,
    const float* __restrict__ vj_all, const float* __restrict__ coords,
    const int* __restrict__ idx,
    const float* __restrict__ gp_w, const float* __restrict__ gp_b,
    const float* __restrict__ gm1_w, const float* __restrict__ gm1_b,
    const float* __restrict__ gm2_w, const float* __restrict__ gm2_b,
    const float* __restrict__ W_w,
    const float* __restrict__ lng, const float* __restrict__ lnb,
    float* __restrict__ vnew) {
  __shared__ float sGm1[64 * 64];   // paired layout (WMMA B)
  __shared__ float sW[64 * 64];     // plain layout (VALU matvec)
  __shared__ float sGpb[4 * 64];    // paired layout, single k-tile
  __shared__ float sGm2[64];
  __shared__ float sGm1b[64];
  __shared__ float sLg[64];
  __shared__ float sLb[64];
  __shared__ float sH[MU_WAVES][16 * 64];  // per-wave transpose staging

  int tid = threadIdx.x;
  stage_w_pairs(gm1_w, sGm1, tid, MU_WAVES * 32);
  for (int k = tid; k < 4096; k += MU_WAVES * 32) sW[k] = W_w[k];
  for (int k = tid; k < 64; k += MU_WAVES * 32) {
    sGm2[k] = gm2_w[k]; sGm1b[k] = gm1_b[k];
    sLg[k] = lng[k];    sLb[k] = lnb[k];
  }
  // gp rows 0..2 = gp_w, row 3 = gp_b; paired: dst = ((h*64 + c)*2 + lohi)
  for (int k = tid; k < 256; k += MU_WAVES * 32) {
    int r = k >> 6, c = k & 63;
    int h = r >> 1, lohi = r & 1;
    float val = (r < 3) ? gp_w[r * 64 + c] : gp_b[c];
    sGpb[((h << 6) + c) * 2 + lohi] = val;
  }
  __syncthreads();

  int wave = tid >> 5, lane = tid & 31;
  int ln = lane & 15, hi = lane >> 4;
  int i = blockIdx.x * MU_WAVES + wave;  // point handled by this wave
  float g2b = gm2_b[0];

  // neighbor indices: own-half rows (m = 8*hi + r) and opposite-half rows
  int nbh[8], nbo[8];
#pragma unroll
  for (int r = 0; r < 8; ++r) {
    nbh[r] = idx[(size_t)i * 16 + 8 * hi + r];
    nbo[r] = idx[(size_t)i * 16 + 8 * (1 - hi) + r];
  }
  int nbl = idx[(size_t)i * 16 + ln];  // neighbor for A-layout row m = ln

  // ---- C init in C-layout: vi[i][n] + vj[nbr(m)][n]
  v8f acc[4];
#pragma unroll
  for (int nt = 0; nt < 4; ++nt) {
    float viv = vi_all[(size_t)i * 64 + nt * 16 + ln];
#pragma unroll
    for (int r = 0; r < 8; ++r)
      acc[nt][r] = viv + vj_all[(size_t)nbh[r] * 64 + nt * 16 + ln];
  }

  // ---- pos_proj via one WMMA per n-tile: A = [rel_pos | 1], B = [gp_w ; gp_b]
  float pix = coords[(size_t)i * 3 + 0], piy = coords[(size_t)i * 3 + 1],
        piz = coords[(size_t)i * 3 + 2];
  float cx = coords[(size_t)nbl * 3 + 0], cy = coords[(size_t)nbl * 3 + 1],
        cz = coords[(size_t)nbl * 3 + 2];
  v2f ap;
  ap.x = hi ? (cz - piz) : (cx - pix);
  ap.y = hi ? 1.0f : (cy - piy);
#pragma unroll
  for (int nt = 0; nt < 4; ++nt) {
    v2f bp = *(const v2f*)&sGpb[((hi << 6) + nt * 16 + ln) * 2];
    acc[nt] = wmma4(ap, bp, acc[nt]);
  }

  // ---- gelu #1, stage h into LDS (C-layout -> A-layout transpose)
#pragma unroll
  for (int nt = 0; nt < 4; ++nt)
#pragma unroll
    for (int r = 0; r < 8; ++r)
      sH[wave][(r + 8 * hi) * 64 + nt * 16 + ln] = gelu_exact(acc[nt][r]);
  __syncthreads();

  v2f af[16];
#pragma unroll
  for (int kk = 0; kk < 16; ++kk)
    af[kk] = *(const v2f*)&sH[wave][ln * 64 + kk * 4 + 2 * hi];

  // ---- h @ gm1_w + gm1_b (64 WMMAs)
  v8f acc2[4];
#pragma unroll
  for (int nt = 0; nt < 4; ++nt) {
    float bv = sGm1b[nt * 16 + ln];
#pragma unroll
    for (int r = 0; r < 8; ++r) acc2[nt][r] = bv;
#pragma unroll
    for (int kk = 0; kk < 16; ++kk)
      acc2[nt] = wmma4(af[kk], bfrag(sGm1, kk, hi, nt * 16 + ln), acc2[nt]);
  }

  // ---- gelu #2 then gate: G[m] = sigmoid(sum_n h2[m][n]*gm2_w[n] + b)
  float p[8];
#pragma unroll
  for (int r = 0; r < 8; ++r) p[r] = 0.f;
#pragma unroll
  for (int nt = 0; nt < 4; ++nt) {
    float wv = sGm2[nt * 16 + ln];
#pragma unroll
    for (int r = 0; r < 8; ++r) p[r] += gelu_exact(acc2[nt][r]) * wv;
  }
#pragma unroll
  for (int off = 1; off < 16; off <<= 1)
#pragma unroll
    for (int r = 0; r < 8; ++r) p[r] += __shfl_xor(p[r], off, 32);
  float G[8], Go[8];
#pragma unroll
  for (int r = 0; r < 8; ++r) {
    G[r]  = sigmoidf(p[r] + g2b);
    Go[r] = __shfl_xor(G[r], 16, 32);  // gates of the other 8 rows
  }

  // ---- integral over 16 neighbors for columns c0=lane, c1=lane+32
  int c0 = lane, c1 = lane + 32;
  float s0 = 0.f, s1 = 0.f;
#pragma unroll
  for (int r = 0; r < 8; ++r) {
    const float* vh = vold + (size_t)nbh[r] * 64;
    const float* vo = vold + (size_t)nbo[r] * 64;
    s0 += G[r] * vh[c0] + Go[r] * vo[c0];
    s1 += G[r] * vh[c1] + Go[r] * vo[c1];
  }
  s0 *= (1.f / 16.f);
  s1 *= (1.f / 16.f);

  // ---- fused residual matvec v[i] @ W_w, ReLU, LayerNorm (in-register)
  const float* vrow = vold + (size_t)i * 64;
  float u0 = s0, u1 = s1;
#pragma unroll 8
  for (int k = 0; k < 64; ++k) {
    float vk = vrow[k];
    u0 += vk * sW[k * 64 + c0];
    u1 += vk * sW[k * 64 + c1];
  }
  u0 = fmaxf(u0, 0.f);
  u1 = fmaxf(u1, 0.f);
  float sum = u0 + u1, sq = u0 * u0 + u1 * u1;
#pragma unroll
  for (int off = 1; off < 32; off <<= 1) {
    sum += __shfl_xor(sum, off, 32);
    sq  += __shfl_xor(sq,  off, 32);
  }
  float mean = sum * (1.f / 64.f);
  float var  = sq * (1.f / 64.f) - mean * mean;
  float inv  = rsqrtf(var + 1e-5f);
  vnew[(size_t)i * 64 + c0] = (u0 - mean) * inv * sLg[c0] + sLb[c0];
  vnew[(size_t)i * 64 + c1] = (u1 - mean) * inv * sLg[c1] + sLb[c1];
}

// ---------------------------------------------------------------------------
// 4) strided argmax of ||v|| over chunks of 128 -> anchor (superpoint) ids
// ---------------------------------------------------------------------------
__global__ void k_anchor_select(const float* __restrict__ v,
                                int* __restrict__ top_ws,
                                float* __restrict__ top_out) {
  __shared__ float sv[CHUNK];
  __shared__ int   si[CHUNK];
  int tid = threadIdx.x;
  int i = blockIdx.x * CHUNK + tid;
  const float* row = v + (size_t)i * 64;
  float ss = 0.f;
#pragma unroll 8
  for (int k = 0; k < 64; ++k) { float x = row[k]; ss += x * x; }
  sv[tid] = ss;
  si[tid] = tid;
  __syncthreads();
  for (int s = CHUNK / 2; s > 0; s >>= 1) {
    if (tid < s) {
      float ov = sv[tid + s]; int oi = si[tid + s];
      if (ov > sv[tid] || (ov == sv[tid] && oi < si[tid])) { sv[tid] = ov; si[tid] = oi; }
    }
    __syncthreads();
  }
  if (tid == 0) {
    int g = blockIdx.x * CHUNK + si[0];
    top_ws[blockIdx.x]  = g;
    top_out[blockIdx.x] = (float)g;
  }
}

// ---------------------------------------------------------------------------
// 5) generic [rows,64]x[64,64] GEMM with optional row gather and output scale
//    (keys = (v@ak+b)/s ; queries = (v[top]@aq+b)/s)
// ---------------------------------------------------------------------------
#define GM_WAVES 8
__global__ void k_gemm64(const float* __restrict__ A, const float* __restrict__ W,
                         const float* __restrict__ bias,
                         const int* __restrict__ rowidx, float scale,
                         float* __restrict__ out) {
  __shared__ float sW[4096];   // paired layout
  int tid = threadIdx.x;
  stage_w_pairs(W, sW, tid, GM_WAVES * 32);
  __syncthreads();

  int wave = tid >> 5, lane = tid & 31;
  int ln = lane & 15, hi = lane >> 4;
  int row0 = (blockIdx.x * GM_WAVES + wave) * 16;
  int arow = row0 + ln;
  if (rowidx) arow = rowidx[arow];

  v2f af[16];
#pragma unroll
  for (int kk = 0; kk < 16; ++kk)
    af[kk] = *(const v2f*)(A + (size_t)arow * 64 + kk * 4 + 2 * hi);

  v8f acc[4];
#pragma unroll
  for (int nt = 0; nt < 4; ++nt) {
    float bv = bias[nt * 16 + ln];
#pragma unroll
    for (int r = 0; r < 8; ++r) acc[nt][r] = bv;
#pragma unroll
    for (int kk = 0; kk < 16; ++kk)
      acc[nt] = wmma4(af[kk], bfrag(sW, kk, hi, nt * 16 + ln), acc[nt]);
  }
#pragma unroll
  for (int nt = 0; nt < 4; ++nt)
#pragma unroll
    for (int r = 0; r < 8; ++r)
      out[(size_t)(row0 + r + 8 * hi) * 64 + nt * 16 + ln] = acc[nt][r] * scale;
}

// ---------------------------------------------------------------------------
// 6) per-point top-8 anchors by distance + softmax attention, 1 wave/point
// ---------------------------------------------------------------------------
__global__ void k_attn(const float* __restrict__ coords,
                       const int* __restrict__ top_ws,
                       const float* __restrict__ keys,
                       const float* __restrict__ queries,
                       float* __restrict__ outA, float* __restrict__ outNN) {
  __shared__ float sA[NSP * 3];
  int tid = threadIdx.x;
  for (int j = tid; j < NSP; j += 256) {
    int t = top_ws[j];
    sA[j * 3 + 0] = coords[(size_t)t * 3 + 0];
    sA[j * 3 + 1] = coords[(size_t)t * 3 + 1];
    sA[j * 3 + 2] = coords[(size_t)t * 3 + 2];
  }
  __syncthreads();

  int wave = tid >> 5, lane = tid & 31;
  int i = blockIdx.x * 8 + wave;
  float px = coords[(size_t)i * 3 + 0], py = coords[(size_t)i * 3 + 1],
        pz = coords[(size_t)i * 3 + 2];

  float d[16];  // distances to anchors lane + 32*jj
#pragma unroll
  for (int jj = 0; jj < 16; ++jj) {
    int a = lane + 32 * jj;
    float dx = sA[a * 3 + 0] - px, dy = sA[a * 3 + 1] - py, dz = sA[a * 3 + 2] - pz;
    d[jj] = dx * dx + dy * dy + dz * dz;
  }

  float kreg0 = keys[(size_t)i * 64 + lane];
  float kreg1 = keys[(size_t)i * 64 + lane + 32];

  int ann[KA];
  float logit[KA];
#pragma unroll
  for (int ka = 0; ka < KA; ++ka) {
    // lane-local min with smallest-index tie-break
    float bv = 3.0e38f;
    int bi = 0x7fffffff;
#pragma unroll
    for (int jj = 0; jj < 16; ++jj) {
      int cand = lane + 32 * jj;
      if (d[jj] < bv || (d[jj] == bv && cand < bi)) { bv = d[jj]; bi = cand; }
    }
    // cross-lane min (all lanes converge to the winner)
#pragma unroll
    for (int off = 16; off > 0; off >>= 1) {
      float ov = __shfl_xor(bv, off, 32);
      int oi = __shfl_xor(bi, off, 32);
      if (ov < bv || (ov == bv && oi < bi)) { bv = ov; bi = oi; }
    }
    ann[ka] = bi;
    // mask the chosen anchor in its owner lane
#pragma unroll
    for (int jj = 0; jj < 16; ++jj)
      if ((bi & 31) == lane && (bi >> 5) == jj) d[jj] = 3.0e38f;
    // logit = dot(keys/s, queries[bi]/s) over 64 channels
    const float* q = queries + (size_t)bi * 64;
    float pl = kreg0 * q[lane] + kreg1 * q[lane + 32];
#pragma unroll
    for (int off = 16; off > 0; off >>= 1) pl += __shfl_xor(pl, off, 32);
    logit[ka] = pl;
  }

  float mx = logit[0];
#pragma unroll
  for (int ka = 1; ka < KA; ++ka) mx = fmaxf(mx, logit[ka]);
  float e[KA], ssum = 0.f;
#pragma unroll
  for (int ka = 0; ka < KA; ++ka) { e[ka] = expf(logit[ka] - mx); ssum += e[ka]; }
  float invs = 1.f / ssum;
  if (lane == 0) {
#pragma unroll
    for (int ka = 0; ka < KA; ++ka) {
      outA[(size_t)i * KA + ka]  = e[ka] * invs;
      outNN[(size_t)i * KA + ka] = (float)ann[ka];
    }
  }
}

// ---------------------------------------------------------------------------
// host launcher
// ---------------------------------------------------------------------------
extern "C" void kernel_launch(void* const* d_in, const int* in_sizes, int n_in,
                              void* d_out, int out_size, void* d_ws, size_t ws_size,
                              hipStream_t stream) {
  (void)in_sizes; (void)n_in; (void)out_size; (void)ws_size;
  const float* coords = (const float*)d_in[0];
  const float* feat   = (const float*)d_in[1];
  const float* lift_w = (const float*)d_in[2];
  const float* lift_b = (const float*)d_in[3];
  const float* gp_w   = (const float*)d_in[4];
  const float* gp_b   = (const float*)d_in[5];
  const float* gvi_w  = (const float*)d_in[6];
  const float* gvi_b  = (const float*)d_in[7];
  const float* gvj_w  = (const float*)d_in[8];
  const float* gm1_w  = (const float*)d_in[9];
  const float* gm1_b  = (const float*)d_in[10];
  const float* gm2_w  = (const float*)d_in[11];
  const float* gm2_b  = (const float*)d_in[12];
  const float* W_w    = (const float*)d_in[13];
  const float* ln_g   = (const float*)d_in[14];
  const float* ln_b   = (const float*)d_in[15];
  const float* ak_w   = (const float*)d_in[16];
  const float* ak_b   = (const float*)d_in[17];
  const float* aq_w   = (const float*)d_in[18];
  const float* aq_b   = (const float*)d_in[19];
  const int*   idx    = (const int*)d_in[20];

  char* ws = (char*)d_ws;
  const size_t SZ = (size_t)N_PTS * HD * sizeof(float);  // 16 MB
  float* vA     = (float*)(ws + 0 * SZ);
  float* vB     = (float*)(ws + 1 * SZ);
  float* vi_all = (float*)(ws + 2 * SZ);
  float* vj_all = (float*)(ws + 3 * SZ);
  int*   top_ws = (int*)(ws + 4 * SZ);
  float* keysb  = vi_all;  // reused after last step
  float* qb     = vj_all;  // reused after last step

  float* outA  = (float*)d_out;                       // [N, KA]
  float* outNN = outA + (size_t)N_PTS * KA;           // [N, KA] (float-cast ints)
  float* outTI = outNN + (size_t)N_PTS * KA;          // [NSP]
  float* outV  = outTI + NSP;                         // [N, H]

  k_lift<<<(N_PTS * HD) / 256, 256, 0, stream>>>(coords, feat, lift_w, lift_b, vA);

  float* vc = vA;
  float* vn = vB;
  for (int t = 0; t < T_STEPS; ++t) {
    k_dual_gemm<<<N_PTS / (DG_WAVES * 16), DG_WAVES * 32, 0, stream>>>(
        vc, gvi_w, gvi_b, gvj_w, vi_all, vj_all);
    k_message_update<<<N_PTS / MU_WAVES, MU_WAVES * 32, 0, stream>>>(
        vc, vi_all, vj_all, coords, idx, gp_w, gp_b, gm1_w, gm1_b, gm2_w, gm2_b,
        W_w, ln_g + t * 64, ln_b + t * 64, vn);
    float* tmp = vc; vc = vn; vn = tmp;
  }

  k_anchor_select<<<NSP, CHUNK, 0, stream>>>(vc, top_ws, outTI);

  const float inv_scale = 0.35355339059327373f;  // 1 / 64^0.25
  k_gemm64<<<N_PTS / (GM_WAVES * 16), GM_WAVES * 32, 0, stream>>>(
      vc, ak_w, ak_b, nullptr, inv_scale, keysb);
  k_gemm64<<<NSP / (GM_WAVES * 16), GM_WAVES * 32, 0, stream>>>(
      vc, aq_w, aq_b, top_ws, inv_scale, qb);

  k_attn<<<N_PTS / 8, 256, 0, stream>>>(coords, top_ws, keysb, qb, outA, outNN);

  hipMemcpyAsync(outV, vc, SZ, hipMemcpyDeviceToDevice, stream);
}